// QuantumConv2DLayer_81003083202958
// MI455X (gfx1250) — compile-verified
//
#include <hip/hip_runtime.h>

// QuantumConv2DLayer collapses to: val = sum(W2 @ relu(W1[:,0]+b1) + b2),
// broadcast over [8,64,224,224]. Scalar via f32 WMMA (exact), then a B128 fill.

#define DIMQ 65536   // 2^NQ state vector length (W1 row stride)
#define HID  256
#define OC   64

typedef __attribute__((ext_vector_type(2))) float v2f;
typedef __attribute__((ext_vector_type(8))) float v8f;

__global__ __launch_bounds__(32)
void qc_scalar_kernel(const float* __restrict__ W1,
                      const float* __restrict__ b1,
                      const float* __restrict__ W2,
                      const float* __restrict__ b2,
                      float* __restrict__ val_out)
{
    __shared__ float hidden[HID];
    const int lane = threadIdx.x;            // 0..31, one wave32

    // hidden = relu(W1[:,0] + b1)  — W1 is [HID, DIMQ] row-major, column 0
    #pragma unroll
    for (int i = 0; i < HID / 32; ++i) {
        int h = lane + i * 32;
        float v = W1[(size_t)h * DIMQ] + b1[h];
        hidden[h] = v > 0.0f ? v : 0.0f;
    }
    __syncthreads();

    // GEMV y = W2 @ hidden via V_WMMA_F32_16X16X4_F32 (exact f32 path).
    // A 16x4 layout: lanes 0-15 -> K = {k, k+1}; lanes 16-31 -> K = {k+2, k+3}.
    // B 4x16 layout mirrors over N; replicate hidden[k..] across all columns.
    const int m    = lane & 15;
    const int kadd = (lane >> 4) * 2;

    float total = 0.0f;
    #pragma unroll
    for (int mt = 0; mt < OC / 16; ++mt) {
        v8f c = {0.f, 0.f, 0.f, 0.f, 0.f, 0.f, 0.f, 0.f};
        const float* __restrict__ arow = W2 + (size_t)(mt * 16 + m) * HID;
        for (int k = 0; k < HID; k += 4) {
            v2f a, b;
            a.x = arow[k + kadd];
            a.y = arow[k + kadd + 1];
            b.x = hidden[k + kadd];
            b.y = hidden[k + kadd + 1];
            // (neg_a, A, neg_b, B, c_mod, C, reuse_a, reuse_b)
            c = __builtin_amdgcn_wmma_f32_16x16x4_f32(
                    false, a, false, b, (short)0, c, false, false);
        }
        // C/D layout: VGPR j holds rows j (lanes 0-15) and j+8 (lanes 16-31);
        // lane index = column. Column 0 lives in lanes 0 and 16.
        float s = c[0] + c[1] + c[2] + c[3] + c[4] + c[5] + c[6] + c[7];
        total += __shfl(s, 0, 32) + __shfl(s, 16, 32);
    }

    // + sum(b2) over OC=64
    float t = b2[lane] + b2[lane + 32];
    #pragma unroll
    for (int off = 16; off > 0; off >>= 1)
        t += __shfl_xor(t, off, 32);

    if (lane == 0)
        val_out[0] = total + t;
}

// Broadcast fill: 102.8 MB of stores, HBM-bound (~4.4 us at 23.3 TB/s).
// float4 stores -> global_store_b128.
__global__ __launch_bounds__(256)
void qc_fill_kernel(const float* __restrict__ valp,
                    float4* __restrict__ out, int n4)
{
    int i = blockIdx.x * blockDim.x + threadIdx.x;
    float v = *valp;   // uniform broadcast read, L2-resident
    if (i < n4)
        out[i] = make_float4(v, v, v, v);
}

extern "C" void kernel_launch(void* const* d_in, const int* in_sizes, int n_in,
                              void* d_out, int out_size, void* d_ws, size_t ws_size,
                              hipStream_t stream)
{
    // setup_inputs order: x, W1, b1, W2, b2, params
    const float* W1 = (const float*)d_in[1];
    const float* b1 = (const float*)d_in[2];
    const float* W2 = (const float*)d_in[3];
    const float* b2 = (const float*)d_in[4];
    float* val = (float*)d_ws;

    qc_scalar_kernel<<<1, 32, 0, stream>>>(W1, b1, W2, b2, val);

    // out_size = 8*64*224*224, divisible by 4
    int n4 = out_size / 4;
    int blocks = (n4 + 255) / 256;
    qc_fill_kernel<<<blocks, 256, 0, stream>>>(val, (float4*)d_out, n4);
}